// SK_PointNet_26173530702025
// MI455X (gfx1250) — compile-verified
//
#include <hip/hip_runtime.h>
#include <math.h>

#define NPTS 4096
#define BATCH 2
#define NBR 2      // branches
#define MAXS 64    // max nsample
#define R2 0.36f

typedef float v2f __attribute__((ext_vector_type(2)));
typedef float v8f __attribute__((ext_vector_type(8)));
typedef unsigned int u32x4 __attribute__((ext_vector_type(4)));
typedef int i32x4 __attribute__((ext_vector_type(4)));
typedef int i32x8 __attribute__((ext_vector_type(8)));

#if defined(__has_builtin)
#if __has_builtin(__builtin_amdgcn_tensor_load_to_lds) && \
    __has_builtin(__builtin_amdgcn_s_wait_tensorcnt)
#define HAVE_TDM 1
#endif
#endif

// ---------------------------------------------------------------------------
// Stage x[b] (NPTS*3 floats = 48KB) into LDS.  On CDNA5 use the Tensor Data
// Mover: one wave issues a 1-D tile descriptor (data_size=4B, 12288 elems),
// waits on TENSORcnt, and the workgroup barrier publishes the LDS contents.
// xs is the kernel's only __shared__ object -> LDS offset 0.
// ---------------------------------------------------------------------------
__device__ __forceinline__ void stage_x_lds(float* xs, const float* __restrict__ xb) {
#ifdef HAVE_TDM
    if (threadIdx.x < 32) {
        const unsigned long long ga = (unsigned long long)(const void*)xb;
        const unsigned nelem = NPTS * 3;          // 12288 f32 elements
        u32x4 g0;
        g0.x = 1u;                                 // count=1, user descriptor
        g0.y = 0u;                                 // lds_addr (xs @ offset 0)
        g0.z = (unsigned)(ga & 0xffffffffu);       // global_addr[31:0]
        g0.w = (unsigned)((ga >> 32) & 0x1ffffffu) // global_addr[56:32]
             | (2u << 30);                         // type = 2 ("image")
        i32x8 g1;
        g1[0] = (int)(2u << 16);                   // data_size = 4 bytes
        g1[1] = (int)((nelem & 0xffffu) << 16);    // tensor_dim0[15:0]
        g1[2] = (int)(((nelem >> 16) & 0xffffu) | (1u << 16)); // dim0 hi | tensor_dim1=1
        g1[3] = (int)((nelem & 0xffffu) << 16);    // tile_dim0 = nelem
        g1[4] = 1;                                 // tile_dim1 = 1
        g1[5] = (int)nelem;                        // tensor_dim0_stride
        g1[6] = 0;
        g1[7] = 0;
        i32x4 gz = {0, 0, 0, 0};                   // groups 2/3 unused (<=2D)
#if __clang_major__ >= 23
        i32x8 gz8 = {0, 0, 0, 0, 0, 0, 0, 0};
        __builtin_amdgcn_tensor_load_to_lds(g0, g1, gz, gz, gz8, 0);
#else
        __builtin_amdgcn_tensor_load_to_lds(g0, g1, gz, gz, 0);
#endif
        __builtin_amdgcn_s_wait_tensorcnt(0);
    }
#else
    for (int t = threadIdx.x; t < NPTS * 3; t += 256) xs[t] = xb[t];
#endif
    __syncthreads();
    (void)xs;
}

// ---------------------------------------------------------------------------
// Kernel 1: ball query.  One thread per query point, x[b] staged in LDS.
// Produces first up-to-64 in-radius column indices (ascending), padded with
// the first hit (matches sort+clamp semantics of the reference).
// ---------------------------------------------------------------------------
__global__ void bq_kernel(const float* __restrict__ x, int* __restrict__ idx) {
    __shared__ float xs[NPTS * 3];
    const int b  = blockIdx.x / (NPTS / 256);
    const int ib = blockIdx.x % (NPTS / 256);
    stage_x_lds(xs, x + (size_t)b * NPTS * 3);

    const int i = ib * 256 + threadIdx.x;
    const float xi0 = xs[i * 3 + 0], xi1 = xs[i * 3 + 1], xi2 = xs[i * 3 + 2];
    int* out = idx + ((size_t)b * NPTS + i) * MAXS;
    int cnt = 0, first = 0;
    for (int j = 0; j < NPTS; ++j) {
        float d0 = xi0 - xs[j * 3 + 0];
        float d1 = xi1 - xs[j * 3 + 1];
        float d2 = xi2 - xs[j * 3 + 2];
        float d  = d0 * d0 + d1 * d1 + d2 * d2;
        if (d <= R2) {
            if (cnt == 0) first = j;
            out[cnt] = j;
            if (++cnt == MAXS) break;
        }
    }
    for (int s = cnt; s < MAXS; ++s) out[s] = first;
}

// ---------------------------------------------------------------------------
// Kernel 2: GAPNet branch compute. One wave32 per (b, branch, point);
// each lane owns 4 of 128 channels; butterfly shuffle reductions.
// ---------------------------------------------------------------------------
__device__ __forceinline__ float wave_sum(float v) {
    for (int off = 16; off; off >>= 1) v += __shfl_xor(v, off, 32);
    return v;
}
__device__ __forceinline__ float wave_max(float v) {
    for (int off = 16; off; off >>= 1) v = fmaxf(v, __shfl_xor(v, off, 32));
    return v;
}

__global__ void gapnet_kernel(const float* __restrict__ x, const int* __restrict__ idx,
                              const float* __restrict__ bW1, const float* __restrict__ bb1,
                              const float* __restrict__ bW2, const float* __restrict__ bb2,
                              const float* __restrict__ bW3, const float* __restrict__ bb3,
                              const float* __restrict__ bW4, const float* __restrict__ bb4,
                              float* __restrict__ y) {
    __shared__ float xs[NPTS * 3];
    const int gw   = blockIdx.x * 8 + (threadIdx.x >> 5);
    const int lane = threadIdx.x & 31;
    const int b   = gw / (NBR * NPTS);
    const int rem = gw % (NBR * NPTS);
    const int k   = rem / NPTS;
    const int i   = rem % NPTS;

    stage_x_lds(xs, x + (size_t)b * NPTS * 3);

    const int ns = k ? 64 : 32;
    const float* W1 = bW1 + (size_t)k * 128 * 3;
    const float* W2 = bW2 + (size_t)k * 128 * 3;
    const float* W3 = bW3 + (size_t)k * 128;
    const float* W4 = bW4 + (size_t)k * 128;
    const float* B1 = bb1 + (size_t)k * 128;
    const float* B2 = bb2 + (size_t)k * 128;
    const float b3 = bb3[k], b4 = bb4[k];

    const int ob = lane * 4;
    float w1[4][3], w2[4][3], r1[4], r2[4], w3r[4], w4r[4];
#pragma unroll
    for (int c = 0; c < 4; ++c) {
#pragma unroll
        for (int d = 0; d < 3; ++d) {
            w1[c][d] = W1[(ob + c) * 3 + d];
            w2[c][d] = W2[(ob + c) * 3 + d];
        }
        r1[c] = B1[ob + c]; r2[c] = B2[ob + c];
        w3r[c] = W3[ob + c]; w4r[c] = W4[ob + c];
    }

    const float xi0 = xs[i * 3 + 0], xi1 = xs[i * 3 + 1], xi2 = xs[i * 3 + 2];

    // sa = relu(newf . W3 + b3), newf = relu(x @ W1^T + b1)
    float acc = 0.f;
#pragma unroll
    for (int c = 0; c < 4; ++c) {
        float nf = fmaxf(w1[c][0] * xi0 + w1[c][1] * xi1 + w1[c][2] * xi2 + r1[c], 0.f);
        acc += nf * w3r[c];
    }
    const float sa = fmaxf(wave_sum(acc) + b3, 0.f);

    const int* ip = idx + ((size_t)b * NPTS + i) * MAXS;

    // pass A: na_s -> leaky(sa+na) stored 2 per lane
    float keep[2] = { -1e30f, -1e30f };
    for (int s = 0; s < ns; ++s) {
        const int j = ip[s];
        const float e0 = xi0 - xs[j * 3 + 0];
        const float e1 = xi1 - xs[j * 3 + 1];
        const float e2 = xi2 - xs[j * 3 + 2];
        float a2 = 0.f;
#pragma unroll
        for (int c = 0; c < 4; ++c) {
            float ef = fmaxf(w2[c][0] * e0 + w2[c][1] * e1 + w2[c][2] * e2 + r2[c], 0.f);
            a2 += ef * w4r[c];
        }
        a2 = wave_sum(a2);
        const float na = fmaxf(a2 + b4, 0.f);
        const float t  = sa + na;
        const float lr = t > 0.f ? t : 0.01f * t;
        if ((s & 31) == lane) keep[s >> 5] = lr;
    }

    // softmax over s
    const float m = wave_max(fmaxf(keep[0], keep[1]));
    float e[2];
    e[0] = expf(keep[0] - m);
    e[1] = (ns > 32) ? expf(keep[1] - m) : 0.f;
    const float inv = 1.f / wave_sum(e[0] + e[1]);

    // pass B: vals = sum_s coef_s * edgef_s ; then ELU
    float vals[4] = {0.f, 0.f, 0.f, 0.f};
    for (int s = 0; s < ns; ++s) {
        const float coef = __shfl(e[s >> 5], s & 31, 32) * inv;
        const int j = ip[s];
        const float e0 = xi0 - xs[j * 3 + 0];
        const float e1 = xi1 - xs[j * 3 + 1];
        const float e2 = xi2 - xs[j * 3 + 2];
#pragma unroll
        for (int c = 0; c < 4; ++c) {
            float ef = fmaxf(w2[c][0] * e0 + w2[c][1] * e1 + w2[c][2] * e2 + r2[c], 0.f);
            vals[c] += coef * ef;
        }
    }
    float* yo = y + (((size_t)(b * NBR + k) * NPTS + i) * 128 + ob);
#pragma unroll
    for (int c = 0; c < 4; ++c) {
        float v = vals[c];
        yo[c] = v > 0.f ? v : (expf(v) - 1.f);
    }
}

// ---------------------------------------------------------------------------
// Kernel 3: s[b][o] = (1/N) * sum_k sum_n y[b,k,n,o]
// ---------------------------------------------------------------------------
__global__ void reduce_y_kernel(const float* __restrict__ y, float* __restrict__ sbuf) {
    const int b = blockIdx.x >> 7, o = blockIdx.x & 127;
    __shared__ float red[256];
    float acc = 0.f;
    for (int k = 0; k < NBR; ++k)
        for (int n = threadIdx.x; n < NPTS; n += 256)
            acc += y[((size_t)(b * NBR + k) * NPTS + n) * 128 + o];
    red[threadIdx.x] = acc;
    __syncthreads();
    for (int s = 128; s; s >>= 1) {
        if (threadIdx.x < (unsigned)s) red[threadIdx.x] += red[threadIdx.x + s];
        __syncthreads();
    }
    if (threadIdx.x == 0) sbuf[blockIdx.x] = red[0] * (1.f / NPTS);
}

// ---------------------------------------------------------------------------
// Kernel 4: fusion weights w[b,k,o] = softmax_k(relu(s@Wf1^T+bf1)@Wf2^T+bf2)
// ---------------------------------------------------------------------------
__global__ void fusion_kernel(const float* __restrict__ sbuf,
                              const float* __restrict__ Wf1, const float* __restrict__ bf1,
                              const float* __restrict__ Wf2, const float* __restrict__ bf2,
                              float* __restrict__ w) {
    const int b = blockIdx.x, t = threadIdx.x; // 128 threads
    __shared__ float z[16];
    if (t < 16) {
        float a = bf1[t];
        for (int c = 0; c < 128; ++c) a += sbuf[b * 128 + c] * Wf1[t * 128 + c];
        z[t] = fmaxf(a, 0.f);
    }
    __syncthreads();
    float l0 = bf2[t], l1 = bf2[128 + t];
    for (int j = 0; j < 16; ++j) {
        l0 += z[j] * Wf2[t * 16 + j];
        l1 += z[j] * Wf2[(128 + t) * 16 + j];
    }
    const float mx = fmaxf(l0, l1);
    const float e0 = expf(l0 - mx), e1 = expf(l1 - mx);
    const float inv = 1.f / (e0 + e1);
    w[(b * NBR + 0) * 128 + t] = e0 * inv;
    w[(b * NBR + 1) * 128 + t] = e1 * inv;
}

// ---------------------------------------------------------------------------
// Kernel 5: combine y with weights, write transposed to out channels 256..383
// ---------------------------------------------------------------------------
__global__ void combine_kernel(const float* __restrict__ y, const float* __restrict__ w,
                               float* __restrict__ out) {
    const int tid = blockIdx.x * 256 + threadIdx.x; // B*N*128
    const int o = tid & 127;
    const int n = (tid >> 7) & (NPTS - 1);
    const int b = tid >> 19;
    float v = y[((size_t)(b * NBR + 0) * NPTS + n) * 128 + o] * w[(b * NBR + 0) * 128 + o]
            + y[((size_t)(b * NBR + 1) * NPTS + n) * 128 + o] * w[(b * NBR + 1) * 128 + o];
    out[((size_t)b * 384 + 256 + o) * NPTS + n] = v;
}

// ---------------------------------------------------------------------------
// Kernel 6: first linear layer (K=3): pre1 = x @ Wc1^T + bc1
// ---------------------------------------------------------------------------
__global__ void lin1_kernel(const float* __restrict__ x, const float* __restrict__ Wc1,
                            const float* __restrict__ bc1, float* __restrict__ pre1) {
    const int tid = blockIdx.x * 256 + threadIdx.x; // B*N*64
    const int c = tid & 63;
    const int m = tid >> 6;
    pre1[tid] = bc1[c] + x[m * 3 + 0] * Wc1[c * 3 + 0]
                        + x[m * 3 + 1] * Wc1[c * 3 + 1]
                        + x[m * 3 + 2] * Wc1[c * 3 + 2];
}

// ---------------------------------------------------------------------------
// Kernel 7: group-norm statistics per (b, group); 8 groups.
// stats[(b*8+g)*2] = mean, [+1] = rsqrt(var+eps)
// ---------------------------------------------------------------------------
__global__ void gn_stats_kernel(const float* __restrict__ pre, int C, float* __restrict__ stats) {
    const int b = blockIdx.x >> 3, g = blockIdx.x & 7;
    const int cpg = C >> 3;
    __shared__ float s1[256], s2[256];
    float a1 = 0.f, a2 = 0.f;
    for (int t = threadIdx.x; t < NPTS * cpg; t += 256) {
        const int n = t / cpg;
        const int c = g * cpg + (t % cpg);
        const float v = pre[((size_t)b * NPTS + n) * C + c];
        a1 += v; a2 += v * v;
    }
    s1[threadIdx.x] = a1; s2[threadIdx.x] = a2;
    __syncthreads();
    for (int s = 128; s; s >>= 1) {
        if (threadIdx.x < (unsigned)s) {
            s1[threadIdx.x] += s1[threadIdx.x + s];
            s2[threadIdx.x] += s2[threadIdx.x + s];
        }
        __syncthreads();
    }
    if (threadIdx.x == 0) {
        const float cnt = (float)NPTS * cpg;
        const float mean = s1[0] / cnt;
        const float var  = s2[0] / cnt - mean * mean;
        stats[blockIdx.x * 2 + 0] = mean;
        stats[blockIdx.x * 2 + 1] = rsqrtf(var + 1e-5f);
    }
}

// ---------------------------------------------------------------------------
// Kernel 8: group-norm apply + relu; optionally write transposed into d_out.
// ---------------------------------------------------------------------------
__global__ void gn_apply_kernel(const float* __restrict__ pre, const float* __restrict__ stats,
                                const float* __restrict__ gamma, const float* __restrict__ beta,
                                int C, int transposed, float* __restrict__ out) {
    const int tid = blockIdx.x * 256 + threadIdx.x; // B*N*C
    const int c = tid % C;
    const int n = (tid / C) & (NPTS - 1);
    const int b = tid / (C * NPTS);
    const int g = c / (C >> 3);
    const float mean = stats[(b * 8 + g) * 2 + 0];
    const float inv  = stats[(b * 8 + g) * 2 + 1];
    float v = (pre[tid] - mean) * inv * gamma[c] + beta[c];
    v = fmaxf(v, 0.f);
    if (transposed) out[((size_t)b * 384 + c) * NPTS + n] = v;
    else            out[tid] = v;
}

// ---------------------------------------------------------------------------
// Kernel 9: f32 WMMA GEMM  out[M,Nout] = A[M,K] @ W[Nout,K]^T + bias
// One wave per 16x16 tile via V_WMMA_F32_16X16X4_F32; 8 waves/block -> 16x128.
// A-frag: lane%16 = M row, (lane/16)*2 = K half.  B-frag mirrors with N col.
// ---------------------------------------------------------------------------
__global__ void wmma_gemm_kernel(const float* __restrict__ A, const float* __restrict__ W,
                                 const float* __restrict__ bias, int K, int Nout,
                                 float* __restrict__ out) {
    const int lane = threadIdx.x & 31;
    const int wid  = threadIdx.x >> 5;
    const int m_base = blockIdx.x * 16;
    const int n_base = blockIdx.y * 128 + wid * 16;
    const int row16  = lane & 15;
    const int khalf  = (lane >> 4) * 2;

    // pull this wave's weight rows toward the WGP (global_prefetch_b8)
    __builtin_prefetch(&W[(size_t)(n_base + row16) * K], 0, 3);

    v8f acc = {};
    for (int k0 = 0; k0 < K; k0 += 4) {
        v2f a, bm;
        const float* ap = A + (size_t)(m_base + row16) * K + k0 + khalf;
        const float* bp = W + (size_t)(n_base + row16) * K + k0 + khalf;
        a.x = ap[0]; a.y = ap[1];
        bm.x = bp[0]; bm.y = bp[1];
        acc = __builtin_amdgcn_wmma_f32_16x16x4_f32(false, a, false, bm,
                                                    (short)0, acc, false, false);
    }

    const int nn   = n_base + (lane & 15);
    const int moff = (lane >> 4) * 8;
    const float bv = bias[nn];
#pragma unroll
    for (int r = 0; r < 8; ++r)
        out[(size_t)(m_base + r + moff) * Nout + nn] = acc[r] + bv;
}

// ---------------------------------------------------------------------------
extern "C" void kernel_launch(void* const* d_in, const int* in_sizes, int n_in,
                              void* d_out, int out_size, void* d_ws, size_t ws_size,
                              hipStream_t stream) {
    (void)in_sizes; (void)n_in; (void)out_size; (void)ws_size;
    const float* x   = (const float*)d_in[0];
    const float* bW1 = (const float*)d_in[1];
    const float* bb1 = (const float*)d_in[2];
    const float* bW2 = (const float*)d_in[3];
    const float* bb2 = (const float*)d_in[4];
    const float* bW3 = (const float*)d_in[5];
    const float* bb3 = (const float*)d_in[6];
    const float* bW4 = (const float*)d_in[7];
    const float* bb4 = (const float*)d_in[8];
    const float* Wc1 = (const float*)d_in[9];
    const float* bc1 = (const float*)d_in[10];
    const float* g1  = (const float*)d_in[11];
    const float* be1 = (const float*)d_in[12];
    const float* Wc2 = (const float*)d_in[13];
    const float* bc2 = (const float*)d_in[14];
    const float* g2  = (const float*)d_in[15];
    const float* be2 = (const float*)d_in[16];
    const float* Wc3 = (const float*)d_in[17];
    const float* bc3 = (const float*)d_in[18];
    const float* g3  = (const float*)d_in[19];
    const float* be3 = (const float*)d_in[20];
    const float* Wf1 = (const float*)d_in[21];
    const float* bf1 = (const float*)d_in[22];
    const float* Wf2 = (const float*)d_in[23];
    const float* bf2 = (const float*)d_in[24];
    float* out = (float*)d_out;

    // workspace carving
    char* ws = (char*)d_ws;
    size_t off = 0;
    auto carve = [&](size_t bytes) { char* p = ws + off; off += (bytes + 255) & ~(size_t)255; return p; };
    int*   idx   = (int*)  carve((size_t)BATCH * NPTS * MAXS * 4);
    float* y     = (float*)carve((size_t)BATCH * NBR * NPTS * 128 * 4);
    float* sbuf  = (float*)carve((size_t)BATCH * 128 * 4);
    float* wbuf  = (float*)carve((size_t)BATCH * NBR * 128 * 4);
    float* pre1  = (float*)carve((size_t)BATCH * NPTS * 64 * 4);
    float* h1    = (float*)carve((size_t)BATCH * NPTS * 64 * 4);
    float* pre2  = (float*)carve((size_t)BATCH * NPTS * 128 * 4);
    float* h2    = (float*)carve((size_t)BATCH * NPTS * 128 * 4);
    float* pre3  = (float*)carve((size_t)BATCH * NPTS * 256 * 4);
    float* st1   = (float*)carve((size_t)BATCH * 8 * 2 * 4);
    float* st2   = (float*)carve((size_t)BATCH * 8 * 2 * 4);
    float* st3   = (float*)carve((size_t)BATCH * 8 * 2 * 4);

    const int M = BATCH * NPTS; // 8192

    // --- attention path ---
    bq_kernel<<<BATCH * (NPTS / 256), 256, 0, stream>>>(x, idx);
    gapnet_kernel<<<BATCH * NBR * NPTS / 8, 256, 0, stream>>>(
        x, idx, bW1, bb1, bW2, bb2, bW3, bb3, bW4, bb4, y);
    reduce_y_kernel<<<BATCH * 128, 256, 0, stream>>>(y, sbuf);
    fusion_kernel<<<BATCH, 128, 0, stream>>>(sbuf, Wf1, bf1, Wf2, bf2, wbuf);
    combine_kernel<<<BATCH * NPTS * 128 / 256, 256, 0, stream>>>(y, wbuf, out);

    // --- MLP + group-norm path ---
    lin1_kernel<<<M * 64 / 256, 256, 0, stream>>>(x, Wc1, bc1, pre1);
    gn_stats_kernel<<<BATCH * 8, 256, 0, stream>>>(pre1, 64, st1);
    gn_apply_kernel<<<M * 64 / 256, 256, 0, stream>>>(pre1, st1, g1, be1, 64, 0, h1);

    {
        dim3 grid(M / 16, 128 / 128);
        wmma_gemm_kernel<<<grid, 256, 0, stream>>>(h1, Wc2, bc2, 64, 128, pre2);
    }
    gn_stats_kernel<<<BATCH * 8, 256, 0, stream>>>(pre2, 128, st2);
    gn_apply_kernel<<<M * 128 / 256, 256, 0, stream>>>(pre2, st2, g2, be2, 128, 0, h2);

    {
        dim3 grid(M / 16, 256 / 128);
        wmma_gemm_kernel<<<grid, 256, 0, stream>>>(h2, Wc3, bc3, 128, 256, pre3);
    }
    gn_stats_kernel<<<BATCH * 8, 256, 0, stream>>>(pre3, 256, st3);
    gn_apply_kernel<<<M * 256 / 256, 256, 0, stream>>>(pre3, st3, g3, be3, 256, 1, out);
}